// GraphAttention_86088324481340
// MI455X (gfx1250) — compile-verified
//
#include <hip/hip_runtime.h>
#include <math.h>

#define NN 50000
#define NE 800000
#define KDIM 256
#define HH 4
#define DD 128
#define LALPHA 0.2f

typedef float v2f __attribute__((ext_vector_type(2)));
typedef float v8f __attribute__((ext_vector_type(8)));

// ---------------- WMMA GEMM: ft = A(N x 256) * B(256 x 128), all f32 ----------------
// Block = 256 threads = 8 waves. Block covers 16 rows; wave w covers cols [16w,16w+16).
// V_WMMA_F32_16X16X4_F32:  A 16x4 (2 VGPR), B 4x16 (2 VGPR), C/D 16x16 (8 VGPR).
__global__ __launch_bounds__(256)
void gemm_ft_kernel(const float* __restrict__ A, const float* __restrict__ B,
                    float* __restrict__ C)
{
    const int wave = threadIdx.x >> 5;
    const int lane = threadIdx.x & 31;
    const int half = lane >> 4;     // 0: lanes 0-15, 1: lanes 16-31
    const int l16  = lane & 15;
    const int row0 = blockIdx.x * 16;
    const int col0 = wave * 16;

    v8f acc = {0.f, 0.f, 0.f, 0.f, 0.f, 0.f, 0.f, 0.f};
    const float* arow = A + (size_t)(row0 + l16) * KDIM;

    #pragma unroll 8
    for (int k = 0; k < KDIM; k += 4) {
        const int ka = k + 2 * half;            // half-wave K offset per ISA layout
        v2f a, b;
        a.x = arow[ka + 0];
        a.y = arow[ka + 1];
        b.x = B[(size_t)(ka + 0) * DD + col0 + l16];
        b.y = B[(size_t)(ka + 1) * DD + col0 + l16];
        acc = __builtin_amdgcn_wmma_f32_16x16x4_f32(false, a, false, b,
                                                    (short)0, acc, false, false);
    }
    // C/D layout: VGPR r -> M = r + 8*half, N = lane&15
    float* cbase = C + (size_t)row0 * DD + col0 + l16;
    #pragma unroll
    for (int r = 0; r < 8; ++r)
        cbase[(size_t)(r + 8 * half) * DD] = acc[r];
}

// ---------------- wave32 reduction helper ----------------
__device__ __forceinline__ float wave_sum(float v) {
    #pragma unroll
    for (int off = 16; off > 0; off >>= 1) v += __shfl_xor(v, off, 32);
    return v;
}
__device__ __forceinline__ float dot4(float4 a, float4 b) {
    return a.x * b.x + a.y * b.y + a.z * b.z + a.w * b.w;
}
__device__ __forceinline__ float lrelu(float x) { return x > 0.f ? x : LALPHA * x; }

// ---------------- per-node stats: inv L2 norm, ftx, fty (one wave / node) ----------------
__global__ __launch_bounds__(256)
void node_stats_kernel(const float* __restrict__ ft, const float* __restrict__ wx,
                       const float* __restrict__ wy, float* __restrict__ inv_norm,
                       float* __restrict__ ftx, float* __restrict__ fty)
{
    const int node = (blockIdx.x * 256 + threadIdx.x) >> 5;
    const int lane = threadIdx.x & 31;
    if (node >= NN) return;
    const float4 f  = ((const float4*)(ft + (size_t)node * DD))[lane];
    const float4 ax = ((const float4*)wx)[lane];
    const float4 ay = ((const float4*)wy)[lane];
    float s2 = wave_sum(dot4(f, f));
    float sx = wave_sum(dot4(f, ax));
    float sy = wave_sum(dot4(f, ay));
    if (lane == 0) {
        inv_norm[node] = rsqrtf(fmaxf(s2, 1e-12f));
        ftx[node] = sx;
        fty[node] = sy;
    }
}

// ---------------- per-edge: cosine sim + leaky-relu nn logit (one wave / edge) ----------
__global__ __launch_bounds__(256)
void edge_logits_kernel(const float* __restrict__ ft, const float* __restrict__ inv_norm,
                        const float* __restrict__ ftx, const float* __restrict__ fty,
                        const long long* __restrict__ idx,
                        float* __restrict__ sim, float* __restrict__ nnl)
{
    const int e = (blockIdx.x * 256 + threadIdx.x) >> 5;
    const int lane = threadIdx.x & 31;
    if (e >= NE) return;
    const int s = (int)idx[2 * (long long)e + 0];
    const int d = (int)idx[2 * (long long)e + 1];
    const float4 fs = ((const float4*)(ft + (size_t)s * DD))[lane];
    const float4 fd = ((const float4*)(ft + (size_t)d * DD))[lane];
    const float dot = wave_sum(dot4(fs, fd));
    if (lane == 0) {
        sim[e] = dot * inv_norm[s] * inv_norm[d];
        nnl[e] = lrelu(ftx[s] + fty[d]);
    }
}

// ---------------- unsorted segment softmax via atomics ----------------
__device__ __forceinline__ unsigned enc_f(float v) {
    unsigned u = __float_as_uint(v);
    return (u >> 31) ? ~u : (u | 0x80000000u);
}
__device__ __forceinline__ float dec_f(unsigned e) {
    return (e >> 31) ? __uint_as_float(e ^ 0x80000000u) : __uint_as_float(~e);
}

__global__ void seg_init_kernel(unsigned* __restrict__ mx, float* __restrict__ sm, int n) {
    const int i = blockIdx.x * 256 + threadIdx.x;
    if (i < n) { mx[i] = enc_f(-INFINITY); sm[i] = 0.f; }
}
__global__ void seg_max_kernel(const float* __restrict__ vals, const long long* __restrict__ idx,
                               unsigned* __restrict__ mx, int ne) {
    const int e = blockIdx.x * 256 + threadIdx.x;
    if (e < ne) atomicMax(&mx[(int)idx[2 * (long long)e]], enc_f(vals[e]));
}
__global__ void seg_exp_kernel(const float* __restrict__ vals, const long long* __restrict__ idx,
                               const unsigned* __restrict__ mx, float* __restrict__ sm,
                               float* __restrict__ out, int ne) {
    const int e = blockIdx.x * 256 + threadIdx.x;
    if (e >= ne) return;
    const int s = (int)idx[2 * (long long)e];
    const float ex = expf(vals[e] - dec_f(mx[s]));
    out[e] = ex;
    atomicAdd(&sm[s], ex);
}
__global__ void seg_div_kernel(float* __restrict__ out, const long long* __restrict__ idx,
                               const float* __restrict__ sm, int ne) {
    const int e = blockIdx.x * 256 + threadIdx.x;
    if (e < ne) out[e] /= sm[(int)idx[2 * (long long)e]];
}

// ---------------- zero fill ----------------
__global__ void fill0_kernel(float* __restrict__ p, int n) {
    const int i = blockIdx.x * 256 + threadIdx.x;
    if (i < n) p[i] = 0.f;
}

// ---------------- scatter aggregation: vr[src] += p[e] * ft[dst] (one wave / edge) -----
__global__ __launch_bounds__(256)
void aggregate_kernel(const float* __restrict__ p, const long long* __restrict__ idx,
                      const float* __restrict__ ft, float* __restrict__ vr)
{
    const int e = (blockIdx.x * 256 + threadIdx.x) >> 5;
    const int lane = threadIdx.x & 31;
    if (e >= NE) return;
    const int s = (int)idx[2 * (long long)e + 0];
    const int d = (int)idx[2 * (long long)e + 1];
    const float pe = p[e];
    const float4 fd = ((const float4*)(ft + (size_t)d * DD))[lane];
    float* o = vr + (size_t)s * DD + lane * 4;
    atomicAdd(o + 0, pe * fd.x);
    atomicAdd(o + 1, pe * fd.y);
    atomicAdd(o + 2, pe * fd.z);
    atomicAdd(o + 3, pe * fd.w);
}

// ---------------- per-node view mixing + ELU (one wave / node) ----------------
__global__ __launch_bounds__(256)
void combine_kernel(const float* __restrict__ ft, const float* __restrict__ vr1,
                    const float* __restrict__ vr2, const float* __restrict__ vr3,
                    const float* __restrict__ wv, const float* __restrict__ bias,
                    float* __restrict__ out)
{
    const int node = (blockIdx.x * 256 + threadIdx.x) >> 5;
    const int lane = threadIdx.x & 31;
    if (node >= NN) return;
    const float4 wlo = ((const float4*)wv)[lane];          // w_view[0:128]
    const float4 whi = ((const float4*)(wv + DD))[lane];   // w_view[128:256]
    const float4 f  = ((const float4*)(ft  + (size_t)node * DD))[lane];
    const float4 v1 = ((const float4*)(vr1 + (size_t)node * DD))[lane];
    const float4 v2 = ((const float4*)(vr2 + (size_t)node * DD))[lane];
    const float4 v3 = ((const float4*)(vr3 + (size_t)node * DD))[lane];

    const float d0 = wave_sum(dot4(f, wlo));
    const float h0 = wave_sum(dot4(f,  whi));
    const float h1 = wave_sum(dot4(v1, whi));
    const float h2 = wave_sum(dot4(v2, whi));
    const float h3 = wave_sum(dot4(v3, whi));

    // coefs = softmax(exp(leaky_relu(va)))
    const float t0 = expf(lrelu(d0 + h0));
    const float t1 = expf(lrelu(d0 + h1));
    const float t2 = expf(lrelu(d0 + h2));
    const float t3 = expf(lrelu(d0 + h3));
    const float m  = fmaxf(fmaxf(t0, t1), fmaxf(t2, t3));
    const float e0 = expf(t0 - m), e1 = expf(t1 - m), e2 = expf(t2 - m), e3 = expf(t3 - m);
    const float inv = 1.f / (e0 + e1 + e2 + e3);
    const float c0 = e0 * inv, c1 = e1 * inv, c2 = e2 * inv, c3 = e3 * inv;

    const float4 b4 = ((const float4*)bias)[lane];
    float4 o;
    o.x = f.x * c0 + v1.x * c1 + v2.x * c2 + v3.x * c3 + b4.x;
    o.y = f.y * c0 + v1.y * c1 + v2.y * c2 + v3.y * c3 + b4.y;
    o.z = f.z * c0 + v1.z * c1 + v2.z * c2 + v3.z * c3 + b4.z;
    o.w = f.w * c0 + v1.w * c1 + v2.w * c2 + v3.w * c3 + b4.w;
    o.x = o.x > 0.f ? o.x : expf(o.x) - 1.f;
    o.y = o.y > 0.f ? o.y : expf(o.y) - 1.f;
    o.z = o.z > 0.f ? o.z : expf(o.z) - 1.f;
    o.w = o.w > 0.f ? o.w : expf(o.w) - 1.f;
    // out points at this head's 128-col slice; row stride = H*D = 512 floats
    ((float4*)(out + (size_t)node * (HH * DD)))[lane] = o;
}

// ================================================================================
extern "C" void kernel_launch(void* const* d_in, const int* in_sizes, int n_in,
                              void* d_out, int out_size, void* d_ws, size_t ws_size,
                              hipStream_t stream) {
    const float*     features = (const float*)d_in[0];
    const long long* adj      = (const long long*)d_in[1];
    const float*     adj_vals = (const float*)d_in[2];
    const float*     W_trans  = (const float*)d_in[3];
    const float*     w_ftx    = (const float*)d_in[4];
    const float*     w_fty    = (const float*)d_in[5];
    const float*     w_view   = (const float*)d_in[6];
    const float*     bias     = (const float*)d_in[7];
    float*           out      = (float*)d_out;

    // workspace layout (floats)
    float* ws = (float*)d_ws;
    float* ft       = ws;                                 // N*D
    float* vr1      = ft  + (size_t)NN * DD;              // N*D
    float* vr2      = vr1 + (size_t)NN * DD;              // N*D
    float* vr3      = vr2 + (size_t)NN * DD;              // N*D
    float* inv_norm = vr3 + (size_t)NN * DD;              // N
    float* ftx      = inv_norm + NN;                      // N
    float* fty      = ftx + NN;                           // N
    float* seg_sum  = fty + NN;                           // N
    unsigned* seg_max = (unsigned*)(seg_sum + NN);        // N
    float* sim      = (float*)(seg_max + NN);             // E
    float* nnl      = sim + NE;                           // E
    float* nnv      = nnl + NE;                           // E
    float* pbuf     = nnv + NE;                           // E

    const int edgeWaveBlocks = (NE * 32 + 255) / 256;
    const int nodeWaveBlocks = (NN * 32 + 255) / 256;
    const int nBlk  = (NN + 255) / 256;
    const int eBlk  = (NE + 255) / 256;
    const int ndBlk = (NN * DD + 255) / 256;

    for (int h = 0; h < HH; ++h) {
        const float* Wh = W_trans + (size_t)h * KDIM * DD;
        gemm_ft_kernel<<<NN / 16, 256, 0, stream>>>(features, Wh, ft);
        node_stats_kernel<<<nodeWaveBlocks, 256, 0, stream>>>(
            ft, w_ftx + (size_t)h * DD, w_fty + (size_t)h * DD, inv_norm, ftx, fty);
        edge_logits_kernel<<<edgeWaveBlocks, 256, 0, stream>>>(
            ft, inv_norm, ftx, fty, adj, sim, nnl);

        // nn_vals = seg_softmax(nnl) keyed by src
        seg_init_kernel<<<nBlk, 256, 0, stream>>>(seg_max, seg_sum, NN);
        seg_max_kernel<<<eBlk, 256, 0, stream>>>(nnl, adj, seg_max, NE);
        seg_exp_kernel<<<eBlk, 256, 0, stream>>>(nnl, adj, seg_max, seg_sum, nnv, NE);
        seg_div_kernel<<<eBlk, 256, 0, stream>>>(nnv, adj, seg_sum, NE);

        const float* viewvals[3] = {adj_vals, sim, nnv};
        float*       vrs[3]      = {vr1, vr2, vr3};
        for (int v = 0; v < 3; ++v) {
            seg_init_kernel<<<nBlk, 256, 0, stream>>>(seg_max, seg_sum, NN);
            seg_max_kernel<<<eBlk, 256, 0, stream>>>(viewvals[v], adj, seg_max, NE);
            seg_exp_kernel<<<eBlk, 256, 0, stream>>>(viewvals[v], adj, seg_max, seg_sum, pbuf, NE);
            seg_div_kernel<<<eBlk, 256, 0, stream>>>(pbuf, adj, seg_sum, NE);
            fill0_kernel<<<ndBlk, 256, 0, stream>>>(vrs[v], NN * DD);
            aggregate_kernel<<<edgeWaveBlocks, 256, 0, stream>>>(pbuf, adj, ft, vrs[v]);
        }

        combine_kernel<<<nodeWaveBlocks, 256, 0, stream>>>(
            ft, vr1, vr2, vr3, w_view + (size_t)h * 2 * DD, bias + (size_t)h * DD,
            out + (size_t)h * DD);
    }
}